// NegativeSamplingLinear_24799141167619
// MI455X (gfx1250) — compile-verified
//
#include <hip/hip_runtime.h>

typedef __attribute__((ext_vector_type(2))) float v2f;
typedef __attribute__((ext_vector_type(8))) float v8f;

#define DDIM  512
#define KNEG  512
#define NOUT  513   // 1 positive + 512 negatives
#define NTILES 33   // ceil(513/16)

__global__ __launch_bounds__(256)
void ns_linear_wmma_kernel(const float* __restrict__ x,
                           const int*   __restrict__ y,
                           const int*   __restrict__ neg_idx,
                           const float* __restrict__ W,
                           float*       __restrict__ out)
{
    __shared__ __align__(16) float xs[DDIM];

    const int b   = blockIdx.x;
    const int tid = threadIdx.x;

    // Stage x[b] (512 f32 = 2KB) into LDS, coalesced: 256 threads x 2 loads.
    xs[tid]       = x[(size_t)b * DDIM + tid];
    xs[tid + 256] = x[(size_t)b * DDIM + tid + 256];
    __syncthreads();

    const int lane = tid & 31;          // wave32
    const int wave = tid >> 5;          // 8 waves per block
    const int m    = lane & 15;         // candidate row within tile (M index)
    const int koff = (lane >> 4) << 1;  // 0 for lanes 0-15, 2 for lanes 16-31

    // Each wave handles tiles of 16 candidate rows: tile j-range [tile*16, tile*16+15]
    for (int tile = wave; tile < NTILES; tile += 8) {
        int j  = tile * 16 + m;
        int jj = (j < NOUT) ? j : 0;    // clamp tail lanes (branchless select)
        int row = (jj == 0) ? y[b] : neg_idx[(size_t)b * KNEG + (jj - 1)];
        const float* wrow = W + (size_t)row * DDIM;

        v8f c = {0.f, 0.f, 0.f, 0.f, 0.f, 0.f, 0.f, 0.f};

        // K loop: 512 features, 4 per WMMA, unrolled 8 => one 128B line per row per iter
        for (int k = 0; k < DDIM; k += 32) {
            v2f a[8], xb[8];
            #pragma unroll
            for (int u = 0; u < 8; ++u) {
                // A 16x4 f32 layout: lane holds row m, K = {k+u*4+koff, +1}
                a[u]  = *(const v2f*)(wrow + k + u * 4 + koff);
                // B 4x16 f32: column n = lane, B[kk][n] = x[k+u*4+kk] (broadcast cols)
                xb[u] = *(const v2f*)(&xs[k + u * 4 + koff]);
            }
            #pragma unroll
            for (int u = 0; u < 8; ++u) {
                c = __builtin_amdgcn_wmma_f32_16x16x4_f32(
                        /*neg_a=*/false, a[u],
                        /*neg_b=*/false, xb[u],
                        /*c_mod=*/(short)0, c,
                        /*reuse_a=*/false, /*reuse_b=*/false);
            }
        }

        // C/D 16x16 f32 layout, column N=0: lane 0 holds M=0..7 in c[0..7],
        // lane 16 holds M=8..15 in c[0..7].
        if (m == 0) {
            int base = tile * 16 + (lane >> 4) * 8;
            #pragma unroll
            for (int i = 0; i < 8; ++i) {
                int jo = base + i;
                if (jo < NOUT) out[(size_t)b * NOUT + jo] = c[i];
            }
        }
    }
}

extern "C" void kernel_launch(void* const* d_in, const int* in_sizes, int n_in,
                              void* d_out, int out_size, void* d_ws, size_t ws_size,
                              hipStream_t stream) {
    const float* x   = (const float*)d_in[0];   // [B, D] f32
    const int*   yy  = (const int*)d_in[1];     // [B] int
    const int*   ni  = (const int*)d_in[2];     // [B, K] int
    const float* W   = (const float*)d_in[3];   // [V, D] f32
    float*       out = (float*)d_out;           // [B, 1+K] f32

    const int B = in_sizes[1];                  // 1024 (y has B elements)
    ns_linear_wmma_kernel<<<B, 256, 0, stream>>>(x, yy, ni, W, out);
}